// proj_splat_32177894981930
// MI455X (gfx1250) — compile-verified
//
#include <hip/hip_runtime.h>

typedef __attribute__((ext_vector_type(2))) float v2f;
typedef __attribute__((ext_vector_type(8))) float v8f;

// ---- problem constants (fixed by reference setup_inputs) ----
constexpr int   NR    = 8;            // batch*view camera rows of feats
constexpr int   FDIM  = 64;
constexpr int   FH    = 64, FW = 64;
constexpr int   BATCH = 2, VIEW = 4;
constexpr int   NVOX  = 48;
constexpr int   NV    = NVOX * NVOX * NVOX;   // 110592
constexpr int   TILES_PER_B = NV / 16;        // 6912
constexpr float RSZ_H = (float)FH / 256.0f;   // 0.25
constexpr float RSZ_W = (float)FW / 256.0f;   // 0.25

// ---------------------------------------------------------------------------
// Stage 1: projection einsum via V_WMMA_F32_16X16X4_F32 (exact f32 MMA, K=4).
// One wave per (batch, 16-voxel tile). A = Kcam[b] (12x4 padded to 16x4),
// B = [gx; gy; gz; 1] (4x16). C (16x16) holds im_p rows for all 4 views.
// ---------------------------------------------------------------------------
__global__ __launch_bounds__(32)
void proj_uv_wmma(const float* __restrict__ Kcam,
                  const float* __restrict__ vmax,
                  const float* __restrict__ vmin,
                  float2* __restrict__ uv)
{
    __shared__ float smem[16 * 16];
    const int lane = threadIdx.x;        // 0..31, wave32
    const int half = lane >> 4;          // lane half selects K pairs
    const int l    = lane & 15;
    const int tile = blockIdx.x;
    const int b    = tile / TILES_PER_B;
    const int n0   = (tile % TILES_PER_B) * 16;

    // ---- B operand: rs_grid[b][:, n0+l] = {gx, gy, gz, 1}
    // Assumed f32 B (4x16) layout: VGPR0 = {K=0 lanes0-15, K=1 lanes16-31},
    //                              VGPR1 = {K=2 lanes0-15, K=3 lanes16-31}.
    const int n  = n0 + l;
    const int ix = n / (NVOX * NVOX);
    const int iy = (n / NVOX) % NVOX;
    const int iz = n % NVOX;
    const float step = 1.0f / (float)(NVOX - 1);
    auto rng = [&](int d, int i) -> float {
        const float lo = vmin[b * 3 + d];
        const float hi = vmax[b * 3 + d];
        return lo + (hi - lo) * ((float)i * step);
    };
    v2f bm;
    bm[0] = half ? rng(1, iy) : rng(0, ix);
    bm[1] = half ? 1.0f       : rng(2, iz);

    // ---- A operand: Kcam[b] rows m = v*3+row (12 valid rows, rest zero).
    // ISA f32 A (16x4) layout: VGPR0 = {K=0 lanes0-15, K=2 lanes16-31},
    //                          VGPR1 = {K=1 lanes0-15, K=3 lanes16-31}.
    const int m  = l;
    const int kb = half * 2;
    v2f am;
    am[0] = (m < VIEW * 3) ? Kcam[(b * 12 + m) * 4 + kb + 0] : 0.0f;
    am[1] = (m < VIEW * 3) ? Kcam[(b * 12 + m) * 4 + kb + 1] : 0.0f;

    v8f c = {};
    c = __builtin_amdgcn_wmma_f32_16x16x4_f32(false, am, false, bm,
                                              (short)0, c, false, false);

    // C layout: lanes 0-15 hold rows 0-7 (VGPR j = row j) of column l,
    //           lanes 16-31 hold rows 8-15. Re-layout through LDS.
    #pragma unroll
    for (int j = 0; j < 8; ++j)
        smem[(half * 8 + j) * 16 + l] = c[j];
    __syncthreads();

    // Each lane finishes two (view, voxel) projections: divide, scale, clip.
    #pragma unroll
    for (int i = 0; i < 2; ++i) {
        const int p   = lane + 32 * i;   // 0..63 = 4 views x 16 cols
        const int v   = p >> 4;
        const int col = p & 15;
        const float px = smem[(3 * v + 0) * 16 + col];
        const float py = smem[(3 * v + 1) * 16 + col];
        const float pz = smem[(3 * v + 2) * 16 + col];
        const float u = fminf(fmaxf(px / pz * RSZ_W, 0.0f), (float)(FW - 1));
        const float w = fminf(fmaxf(py / pz * RSZ_H, 0.0f), (float)(FH - 1));
        uv[(size_t)(b * VIEW + v) * NV + (n0 + col)] = make_float2(u, w);
    }
}

// ---------------------------------------------------------------------------
// Stage 2: gather + bilinear. feats (8 MB) is fully L2-resident; the kernel
// is limited by the 226 MB of coalesced float4 stores (~10 us HBM floor).
// One thread per (r, c, 4 consecutive voxels).
// ---------------------------------------------------------------------------
__global__ __launch_bounds__(256)
void gather_bilinear(const float* __restrict__ feats,
                     const float2* __restrict__ uv,
                     float* __restrict__ out)
{
    constexpr int NQ = NV / 4;                       // 27648 quads per (r,c)
    const long long idx = (long long)blockIdx.x * blockDim.x + threadIdx.x;
    if (idx >= (long long)NR * FDIM * NQ) return;
    const int q  = (int)(idx % NQ);
    const int rc = (int)(idx / NQ);                  // r*FDIM + c
    const int r  = rc / FDIM;
    const int n4 = q * 4;

    // 4 consecutive (u,v) pairs = two aligned float4 loads
    const float4* uvp = reinterpret_cast<const float4*>(uv + (size_t)r * NV + n4);
    const float4 uv01 = uvp[0];
    const float4 uv23 = uvp[1];
    const float us[4] = {uv01.x, uv01.z, uv23.x, uv23.z};
    const float vs[4] = {uv01.y, uv01.w, uv23.y, uv23.w};

    const float* __restrict__ F = feats + (size_t)rc * (FH * FW);
    float4 res;
    float* rp = &res.x;
    #pragma unroll
    for (int k = 0; k < 4; ++k) {
        const float u = us[k], v = vs[k];
        const float x0 = floorf(u), y0 = floorf(v);
        const float x1 = fminf(x0 + 1.0f, (float)(FW - 1));
        const float y1 = fminf(y0 + 1.0f, (float)(FH - 1));
        const int ix0 = (int)x0, iy0 = (int)y0;
        const int ix1 = (int)x1, iy1 = (int)y1;
        // exact reference weights (do NOT assume they sum to 1 at the border)
        const float wx0 = x1 - u, wx1 = u - x0;
        const float wy0 = y1 - v, wy1 = v - y0;
        const float Ia = F[iy0 * FW + ix0];
        const float Ib = F[iy1 * FW + ix0];
        const float Ic = F[iy0 * FW + ix1];
        const float Id = F[iy1 * FW + ix1];
        rp[k] = wx0 * wy0 * Ia + wx0 * wy1 * Ib + wx1 * wy0 * Ic + wx1 * wy1 * Id;
    }
    reinterpret_cast<float4*>(out + (size_t)rc * NV)[q] = res;
}

// ---------------------------------------------------------------------------
extern "C" void kernel_launch(void* const* d_in, const int* in_sizes, int n_in,
                              void* d_out, int out_size, void* d_ws, size_t ws_size,
                              hipStream_t stream)
{
    (void)in_sizes; (void)n_in; (void)out_size; (void)ws_size;
    const float* feats = (const float*)d_in[0];   // (8, 64, 64, 64)
    const float* Kcam  = (const float*)d_in[1];   // (2, 4, 3, 4)
    const float* vmax  = (const float*)d_in[2];   // (2, 3)
    const float* vmin  = (const float*)d_in[3];   // (2, 3)

    float2* uv = (float2*)d_ws;                   // NR*NV float2 = 7.08 MB

    proj_uv_wmma<<<BATCH * TILES_PER_B, 32, 0, stream>>>(Kcam, vmax, vmin, uv);

    const long long total  = (long long)NR * FDIM * (NV / 4);
    const int       blocks = (int)((total + 255) / 256);
    gather_bilinear<<<blocks, 256, 0, stream>>>(feats, uv, (float*)d_out);
}